// CombinedAcceptedRound4Encoder_52355651338331
// MI455X (gfx1250) — compile-verified
//
#include <hip/hip_runtime.h>
#include <math.h>

typedef float v2f __attribute__((ext_vector_type(2)));
typedef float v8f __attribute__((ext_vector_type(8)));

#define TBn 128      // pooled timesteps
#define Rn 64        // resonators
#define Cn 3         // channels
#define TIN 2048     // input timesteps
#define WIN 16       // pooling window (TIN / TBn)
#define DSTRIDE 65   // LDS row stride for drive[t][r] (padded vs 64 banks)

// One block per sample b. 64 threads = 2 wave32s.
// Phase 1: pool traces[b] (24 KB) -> pooled[3][128] in LDS + max-abs reduce.
// Phase 2: drive = (W/scale) x pooled + bias via V_WMMA_F32_16X16X4_F32 -> LDS.
// Phase 3: per-r sequential 128-step spike scan, mean spikes -> out[b][r].
__global__ __launch_bounds__(64) void spike_resonator_fused(
    const float* __restrict__ traces,     // [B,3,2048]
    const float* __restrict__ drive_w,    // [64,3]
    const float* __restrict__ drive_b,    // [64]
    const float* __restrict__ freq_raw,   // [64]
    const float* __restrict__ q_raw,      // [64]
    const float* __restrict__ thr_raw,    // [64]
    float* __restrict__ out)              // [B,64]
{
    __shared__ float sp_pool[Cn * TBn];       // 384 floats, [c][t]
    __shared__ float sdrive[TBn * DSTRIDE];   // 8320 floats, [t][r] padded
    __shared__ float red[64];
    __shared__ float s_inv;

    const int tid = threadIdx.x;
    const int b   = blockIdx.x;
    const float* tb = traces + (size_t)b * (Cn * TIN);

    // ---------- Phase 1: adaptive avg pool (window 16) + max-abs ----------
    float mx = 0.0f;
#pragma unroll
    for (int k = 0; k < 6; ++k) {
        const int p = tid + k * 64;                 // pooled index = c*128+t
        const float4* q4 = (const float4*)(tb + p * WIN);
        float4 x0 = q4[0], x1 = q4[1], x2 = q4[2], x3 = q4[3];
        float s = (x0.x + x0.y + x0.z + x0.w) + (x1.x + x1.y + x1.z + x1.w)
                + (x2.x + x2.y + x2.z + x2.w) + (x3.x + x3.y + x3.z + x3.w);
        s *= (1.0f / 16.0f);
        sp_pool[p] = s;
        mx = fmaxf(mx, fabsf(s));
    }
    red[tid] = mx;
    __syncthreads();
#pragma unroll
    for (int off = 32; off > 0; off >>= 1) {
        if (tid < off) red[tid] = fmaxf(red[tid], red[tid + off]);
        __syncthreads();
    }
    if (tid == 0) s_inv = 1.0f / fmaxf(red[0], 1.0f);
    __syncthreads();

    // ---------- Phase 2: drive GEMM via WMMA f32 16x16x4 (branchless operands) ----------
    {
        const int wave = tid >> 5;      // 0..1 -> r halves 0-31 / 32-63
        const int lane = tid & 31;
        const int hi   = lane >> 4;     // lanes 16-31 hold K=2,3 (K=3 padded 0)
        const int ln   = lane & 15;
        const float inv  = s_inv;
        const float mhi  = hi ? 0.0f : 1.0f;   // zero-pad K=1/K=3 slot in hi half
        const int   wsel = hi ? 2 : 0;         // K=0 vs K=2 weight column
        const int   psel = hi ? 2 * TBn : 0;   // pooled plane c=0 vs c=2
#pragma unroll
        for (int rt = 0; rt < 2; ++rt) {
            const int rtile = wave * 32 + rt * 16;
            const int r = rtile + ln;               // A: M = ln for both halves
            v2f a;
            a[0] = drive_w[r * 3 + wsel] * inv;
            a[1] = drive_w[r * 3 + 1] * (inv * mhi);
            v8f cc;                                 // C = bias broadcast per row M
#pragma unroll
            for (int v = 0; v < 8; ++v) cc[v] = drive_b[rtile + hi * 8 + v];
#pragma unroll
            for (int tt = 0; tt < 8; ++tt) {
                const int t0 = tt * 16;
                v2f bm;                             // B: N = ln, K rows of pooled
                bm[0] = sp_pool[psel + t0 + ln];
                bm[1] = sp_pool[TBn + t0 + ln] * mhi;
                v8f d = __builtin_amdgcn_wmma_f32_16x16x4_f32(
                    false, a, false, bm, (short)0, cc, false, false);
                // D: lane holds column t = t0+ln, rows rtile+8*hi+v (8 contiguous r)
#pragma unroll
                for (int v = 0; v < 8; ++v)
                    sdrive[(t0 + ln) * DSTRIDE + rtile + hi * 8 + v] = d[v];
            }
        }
    }
    __syncthreads();

    // ---------- Phase 3: 128-step sequential spike scan (thread = r) ----------
    {
        const int r = tid;
        const float freq  = 0.03f + 0.17f / (1.0f + expf(-freq_raw[r]));
        const float qf    = 1.5f + log1pf(expf(q_raw[r]));
        const float decay = expf(-1.0f / qf);
        const float thr   = 0.35f + 0.75f / (1.0f + expf(-thr_raw[r]));
        float st = 0.0f, vel = 0.0f, acc = 0.0f;
#pragma unroll 4
        for (int t = 0; t < TBn; ++t) {
            const float cur = sdrive[t * DSTRIDE + r];
            vel = decay * vel + cur - freq * st;
            st += freq * vel;
            const float sp = 1.0f / (1.0f + expf(-10.0f * (st - thr)));
            st -= sp * thr;
            acc += sp;
        }
        out[(size_t)b * Rn + r] = acc * (1.0f / 128.0f);
    }
}

extern "C" void kernel_launch(void* const* d_in, const int* in_sizes, int n_in,
                              void* d_out, int out_size, void* d_ws, size_t ws_size,
                              hipStream_t stream) {
    const float* traces   = (const float*)d_in[0];
    const float* drive_w  = (const float*)d_in[1];
    const float* drive_b  = (const float*)d_in[2];
    const float* freq_raw = (const float*)d_in[3];
    const float* q_raw    = (const float*)d_in[4];
    const float* thr_raw  = (const float*)d_in[5];
    float* out = (float*)d_out;

    const int B = in_sizes[0] / (Cn * TIN);   // 8192
    dim3 grid(B), block(64);
    spike_resonator_fused<<<grid, block, 0, stream>>>(
        traces, drive_w, drive_b, freq_raw, q_raw, thr_raw, out);
}